// VoxelPointNet_51659866636803
// MI455X (gfx1250) — compile-verified
//
#include <hip/hip_runtime.h>

// VoxelPointNet fused kernel for gfx1250 (MI455X), fp32 end-to-end.
// One wave32 processes one voxel (32 points) per grid-stride iteration.
// GEMM1 (32x4 @ 4x16) and GEMM2 (32x16 @ 16x16) run on V_WMMA_F32_16X16X4_F32.
//
// Roofline: ~232 MB total traffic -> ~10 us at 23.3 TB/s; ~8.2 GFLOP fp32.
// Scalar-FMA would be VALU-bound (~8x slower than memory bound); WMMA f32
// (K=4 native shape) moves the GEMM FLOPs to the matrix pipe, leaving only
// the LayerNorms on VALU, which brings the kernel back to bandwidth-bound.

typedef __attribute__((ext_vector_type(2))) float v2f;
typedef __attribute__((ext_vector_type(4))) float v4f;
typedef __attribute__((ext_vector_type(8))) float v8f;

#define LN_EPS 1e-5f
// Dwords per staged LDS row: 20 -> 80B rows, 16B aligned for b128, bank-rotating.
#define ROW_STRIDE 20
#define WAVES_PER_BLOCK 8

__global__ __launch_bounds__(256) void voxel_pointnet_kernel(
    const float* __restrict__ features,   // (V, 32, 4)
    const float* __restrict__ W1,         // (4, 16)
    const float* __restrict__ b1,         // (16,)
    const float* __restrict__ g1,         // (16,)
    const float* __restrict__ be1,        // (16,)
    const float* __restrict__ W2,         // (16, 16)
    const float* __restrict__ b2,         // (16,)
    const float* __restrict__ g2,         // (16,)
    const float* __restrict__ be2,        // (16,)
    const int*   __restrict__ num_points, // (V,)
    float* __restrict__ out,              // (V, 16)
    int V)
{
    __shared__ float smem[WAVES_PER_BLOCK * 32 * ROW_STRIDE];

    const int lane = threadIdx.x & 31;
    const int wave = threadIdx.x >> 5;
    const int half = lane >> 4;     // 0: lanes 0-15, 1: lanes 16-31
    const int l15  = lane & 15;
    float* stage = smem + wave * 32 * ROW_STRIDE;   // wave-private 32x20 dwords

    // ---------------- wave-invariant operands (hoisted) ----------------
    // B-layout for 16x16x4 f32 (K x N): vgpr0 holds K = 2*half, vgpr1 holds K = 2*half+1.
    v2f bw1;
    bw1.x = W1[(half * 2 + 0) * 16 + l15];
    bw1.y = W1[(half * 2 + 1) * 16 + l15];

    v2f bw2[4];                                      // four K-slices of W2 (K=16)
#pragma unroll
    for (int s = 0; s < 4; ++s) {
        bw2[s].x = W2[(4 * s + half * 2 + 0) * 16 + l15];
        bw2[s].y = W2[(4 * s + half * 2 + 1) * 16 + l15];
    }

    const float b1n  = b1[l15];
    const float b2n  = b2[l15];
    const float g2n  = g2[l15];
    const float be2n = be2[l15];

    // LN1 affine params by hidden column (uniform -> scalarized by compiler).
    float g1r[16], be1r[16];
#pragma unroll
    for (int j = 0; j < 16; ++j) { g1r[j] = g1[j]; be1r[j] = be1[j]; }

    const int nwaves = (gridDim.x * blockDim.x) >> 5;
    const int wid    = (blockIdx.x * blockDim.x + threadIdx.x) >> 5;

    for (int v = wid; v < V; v += nwaves) {
        const float* fv = features + (size_t)v * 128;
        // Prefetch next grid-stride voxel with high locality -> WGP-scope
        // prefetch (pulls into near cache, not just GL2).
        if (v + nwaves < V)
            __builtin_prefetch(features + (size_t)(v + nwaves) * 128, 0, 3);

        // ---------------- GEMM1: two 16x4 A-tiles, K=4 in one WMMA each --------
        // A-layout f32 16x4: lane = M + 16*(k>=2), vgpr = k&1 -> direct b64 loads;
        // the wave's 32 x 8B requests cover a contiguous 256B span per tile.
        v2f a0 = *(const v2f*)(fv + (0  + l15) * 4 + half * 2);
        v2f a1 = *(const v2f*)(fv + (16 + l15) * 4 + half * 2);

        v8f c0, c1;
#pragma unroll
        for (int i = 0; i < 8; ++i) { c0[i] = b1n; c1[i] = b1n; }  // C = +b1 broadcast

        c0 = __builtin_amdgcn_wmma_f32_16x16x4_f32(false, a0, false, bw1,
                                                   (short)0, c0, false, false);
        c1 = __builtin_amdgcn_wmma_f32_16x16x4_f32(false, a1, false, bw1,
                                                   (short)0, c1, false, false);

        // Stage H (32x16) into LDS: C-layout element (M,N) -> row M, col N.
#pragma unroll
        for (int i = 0; i < 8; ++i) {
            stage[(0  + i + half * 8) * ROW_STRIDE + l15] = c0[i];
            stage[(16 + i + half * 8) * ROW_STRIDE + l15] = c1[i];
        }
        __builtin_amdgcn_wave_barrier();

        // ---------------- per-point LayerNorm + ReLU + mask (lane = point) -----
        const int np = num_points[v];
        {
            alignas(16) float h[16];
            const float* rowp = stage + lane * ROW_STRIDE;
            v4f* hv = (v4f*)h;
#pragma unroll
            for (int q = 0; q < 4; ++q) hv[q] = *(const v4f*)(rowp + 4 * q);

            float sum = 0.0f, ssq = 0.0f;
#pragma unroll
            for (int j = 0; j < 16; ++j) { sum += h[j]; ssq += h[j] * h[j]; }
            const float mu  = sum * 0.0625f;
            const float var = ssq * 0.0625f - mu * mu;
            const float rs  = __builtin_amdgcn_rsqf(var + LN_EPS);
            const float keep = (lane < np) ? 1.0f : 0.0f;   // fold point mask in
#pragma unroll
            for (int j = 0; j < 16; ++j)
                h[j] = fmaxf((h[j] - mu) * rs * g1r[j] + be1r[j], 0.0f) * keep;

            float* roww = stage + lane * ROW_STRIDE;
#pragma unroll
            for (int q = 0; q < 4; ++q) *(v4f*)(roww + 4 * q) = hv[q];
        }
        __builtin_amdgcn_wave_barrier();

        // ---------------- GEMM2: H(32x16) @ W2(16x16), 4 chained K=4 WMMAs -----
        v8f d0, d1;
#pragma unroll
        for (int i = 0; i < 8; ++i) { d0[i] = 0.0f; d1[i] = 0.0f; }
#pragma unroll
        for (int s = 0; s < 4; ++s) {
            // A-layout slice reads: b64 at row (tile*16 + l15), K = 4s + 2*half.
            v2f a2_0 = *(const v2f*)(stage + (0  + l15) * ROW_STRIDE + 4 * s + half * 2);
            v2f a2_1 = *(const v2f*)(stage + (16 + l15) * ROW_STRIDE + 4 * s + half * 2);
            d0 = __builtin_amdgcn_wmma_f32_16x16x4_f32(false, a2_0, false, bw2[s],
                                                       (short)0, d0, false, false);
            d1 = __builtin_amdgcn_wmma_f32_16x16x4_f32(false, a2_1, false, bw2[s],
                                                       (short)0, d1, false, false);
        }

        // ---------------- masked column sum over 32 points ---------------------
        float col = 0.0f;
#pragma unroll
        for (int i = 0; i < 8; ++i) col += d0[i] + d1[i];   // rows of this half
        col += __shfl_xor(col, 16, 32);                     // + other half's rows
        col += (float)np * b2n;                             // masked +b2 analytically

        // ---------------- final LayerNorm over OUT=16 (16-lane reduction) ------
        float s1 = col, s2 = col * col;
#pragma unroll
        for (int m = 1; m <= 8; m <<= 1) {
            s1 += __shfl_xor(s1, m, 32);
            s2 += __shfl_xor(s2, m, 32);
        }
        const float mu2  = s1 * 0.0625f;
        const float var2 = s2 * 0.0625f - mu2 * mu2;
        const float o = (col - mu2) * __builtin_amdgcn_rsqf(var2 + LN_EPS) * g2n + be2n;

        if (half == 0) out[(size_t)v * 16 + l15] = o;       // 64B coalesced store
    }
}

extern "C" void kernel_launch(void* const* d_in, const int* in_sizes, int n_in,
                              void* d_out, int out_size, void* d_ws, size_t ws_size,
                              hipStream_t stream) {
    const float* features   = (const float*)d_in[0];
    const float* W1         = (const float*)d_in[1];
    const float* b1         = (const float*)d_in[2];
    const float* g1         = (const float*)d_in[3];
    const float* be1        = (const float*)d_in[4];
    const float* W2         = (const float*)d_in[5];
    const float* b2         = (const float*)d_in[6];
    const float* g2         = (const float*)d_in[7];
    const float* be2        = (const float*)d_in[8];
    const int*   num_points = (const int*)d_in[9];
    float* out = (float*)d_out;

    const int V = in_sizes[9];                       // num_points has V elements
    int blocks = (V + WAVES_PER_BLOCK - 1) / WAVES_PER_BLOCK;
    if (blocks > 4096) blocks = 4096;                // grid-stride, ~12 voxels/wave
    if (blocks < 1) blocks = 1;

    voxel_pointnet_kernel<<<blocks, 32 * WAVES_PER_BLOCK, 0, stream>>>(
        features, W1, b1, g1, be1, W2, b2, g2, be2, num_points, out, V);
}